// InvertiblePWL_18116172054655
// MI455X (gfx1250) — compile-verified
//
#include <hip/hip_runtime.h>
#include <math.h>
#include <stdint.h>

#define N_BINS 100
#define PWL_VMIN (-5.0f)
#define PWL_VMAX (5.0f)

// ---------------------------------------------------------------------------
// CDNA5 TDM (tensor_load_to_lds) staging of the small lookup tables into LDS.
// This toolchain exposes the 6-arg (amdgpu-toolchain / clang-23) form:
//   (uint32x4 g0, int32x8 g1, int32x4 g2, int32x4 g3, int32x8 extra, i32 cpol)
// ---------------------------------------------------------------------------
#if defined(__gfx1250__) && __has_builtin(__builtin_amdgcn_tensor_load_to_lds) && \
    __has_builtin(__builtin_amdgcn_s_wait_tensorcnt)
#define USE_TDM 1
#else
#define USE_TDM 0
#endif

#if USE_TDM
typedef unsigned int v4u __attribute__((ext_vector_type(4)));
typedef int          v8i __attribute__((ext_vector_type(8)));
typedef int          v4i __attribute__((ext_vector_type(4)));

// 1D row copy (nelem float32) global -> LDS via the Tensor Data Mover.
// D# group0: count=1, lds_addr, global_addr[56:0], type=2.
// D# group1: data_size=4B, tensor_dim0=nelem, tensor_dim1=1, tile=(nelem,1),
//            dim0_stride=nelem.
__device__ __forceinline__ void tdm_load_row(uint32_t lds_off, const void* gptr,
                                             uint32_t nelem) {
  uint64_t ga = (uint64_t)(uintptr_t)gptr;
  v4u g0;
  g0.x = 1u;                                           // count=1, is_restore=0
  g0.y = lds_off;                                      // lds_addr (bytes)
  g0.z = (uint32_t)(ga & 0xffffffffu);                 // global_addr[31:0]
  g0.w = (uint32_t)((ga >> 32) & 0x01ffffffu) | (2u << 30); // addr[56:32]|type=2
  v8i g1;
  g1[0] = 0x20000;                                     // data_size=2 (4 bytes)
  g1[1] = (int)((nelem & 0xffffu) << 16);              // tensor_dim0[15:0]
  g1[2] = (int)(((nelem >> 16) & 0xffffu) | (1u << 16)); // dim0[31:16]|dim1=1
  g1[3] = (int)((nelem & 0xffffu) << 16);              // dim1 hi=0 | tile_dim0
  g1[4] = 1;                                           // tile_dim1=1, tile_dim2=0
  g1[5] = (int)nelem;                                  // tensor_dim0_stride lo
  g1[6] = 0;
  g1[7] = 0;
  v4i z4 = {0, 0, 0, 0};
  v8i z8 = {0, 0, 0, 0, 0, 0, 0, 0};
  __builtin_amdgcn_tensor_load_to_lds(g0, g1, z4, z4, z8, 0);
}
#endif

// ---------------------------------------------------------------------------
// Prep kernel: k_tab[i] = exp(p[i]) + 1e-3 (i=0..100);
// dbias[j] = b + sum_{i=1..j} h * k_tab[i]  (j=0..99). One tiny block.
// ---------------------------------------------------------------------------
__global__ void pwl_prep_kernel(const float* __restrict__ p,
                                const float* __restrict__ b,
                                float* __restrict__ k_tab,
                                float* __restrict__ dbias) {
  __shared__ float s_k[N_BINS + 1];
  int t = threadIdx.x;
  if (t <= N_BINS) {
    float v = expf(p[t]) + 0.001f;
    s_k[t] = v;
    k_tab[t] = v;
  }
  __syncthreads();
  if (t == 0) {
    const float h = (PWL_VMAX - PWL_VMIN) / (float)(N_BINS - 1);
    float bb = b[0];
    float acc = 0.0f;
    dbias[0] = bb;
    for (int i = 1; i < N_BINS; ++i) {
      acc += h * s_k[i];
      dbias[i] = bb + acc;
    }
  }
}

// ---------------------------------------------------------------------------
// Main streaming kernel.
// idx = #{ j : x >= points[j] } computed analytically (uniform grid) with a
// single BRANCHLESS fixup step against the LDS points table. pts[] carries
// -inf / +inf sentinels at [-1] and [N_BINS] so the fixup compares are always
// in-bounds and trivially false at the clamped extremes.
// ---------------------------------------------------------------------------
__device__ __forceinline__ float pwl_eval(float x, const float* __restrict__ pts,
                                          const float* __restrict__ kt,
                                          const float* __restrict__ db,
                                          float inv_h) {
  int t = (int)floorf((x - PWL_VMIN) * inv_h) + 1;
  t = (t < 0) ? 0 : ((t > N_BINS) ? N_BINS : t);       // v_med3
  t += (x >= pts[t]) ? 1 : 0;    // fp off-by-one up   (pts[N_BINS]=+inf)
  t -= (x < pts[t - 1]) ? 1 : 0; // fp off-by-one down (pts[-1]=-inf)
  int s = t - 1;
  if (s < 0) s = 0;
  return (x - pts[s]) * kt[t] + db[s];
}

__global__ __launch_bounds__(256) void pwl_main_kernel(
    const float* __restrict__ eps, const float* __restrict__ points,
    const float* __restrict__ k_tab, const float* __restrict__ dbias,
    float* __restrict__ out, int n, int n4) {
  __shared__ float s_pts_raw[N_BINS + 2];  // [-inf, points[0..99], +inf]
  __shared__ float s_k[N_BINS + 1];
  __shared__ float s_db[N_BINS];
  float* s_pts = &s_pts_raw[1];

#if USE_TDM
  if (threadIdx.x < 32) {  // wave 0 only (wave32): one TDM issue per table
    tdm_load_row((uint32_t)(uintptr_t)&s_pts[0], points, N_BINS);
    tdm_load_row((uint32_t)(uintptr_t)&s_k[0], k_tab, N_BINS + 1);
    tdm_load_row((uint32_t)(uintptr_t)&s_db[0], dbias, N_BINS);
  }
  if (threadIdx.x == 0) {  // sentinels (disjoint LDS addrs from TDM writes)
    s_pts_raw[0] = -__builtin_huge_valf();
    s_pts_raw[N_BINS + 1] = __builtin_huge_valf();
  }
  if (threadIdx.x < 32) {
    __builtin_amdgcn_s_wait_tensorcnt(0);
  }
  __syncthreads();
#else
  for (int i = threadIdx.x; i < N_BINS; i += blockDim.x) {
    s_pts[i] = points[i];
    s_db[i] = dbias[i];
  }
  for (int i = threadIdx.x; i <= N_BINS; i += blockDim.x) s_k[i] = k_tab[i];
  if (threadIdx.x == 0) {
    s_pts_raw[0] = -__builtin_huge_valf();
    s_pts_raw[N_BINS + 1] = __builtin_huge_valf();
  }
  __syncthreads();
#endif

  const float inv_h = (float)(N_BINS - 1) / (PWL_VMAX - PWL_VMIN);
  const float4* __restrict__ eps4 = (const float4*)eps;
  float4* __restrict__ out4 = (float4*)out;

  int gid = blockIdx.x * blockDim.x + threadIdx.x;
  int stride = gridDim.x * blockDim.x;

  for (int i = gid; i < n4; i += stride) {
    if (i + stride < n4) {
      __builtin_prefetch(&eps4[i + stride], 0, 1);  // global_prefetch_b8
    }
    float4 e = eps4[i];
    float4 r;
    r.x = pwl_eval(e.x, s_pts, s_k, s_db, inv_h);
    r.y = pwl_eval(e.y, s_pts, s_k, s_db, inv_h);
    r.z = pwl_eval(e.z, s_pts, s_k, s_db, inv_h);
    r.w = pwl_eval(e.w, s_pts, s_k, s_db, inv_h);
    out4[i] = r;
  }

  // scalar tail (n not divisible by 4)
  for (int i = 4 * n4 + gid; i < n; i += stride) {
    out[i] = pwl_eval(eps[i], s_pts, s_k, s_db, inv_h);
  }
}

// ---------------------------------------------------------------------------
extern "C" void kernel_launch(void* const* d_in, const int* in_sizes, int n_in,
                              void* d_out, int out_size, void* d_ws,
                              size_t ws_size, hipStream_t stream) {
  const float* eps = (const float*)d_in[0];     // [N]
  const float* p = (const float*)d_in[1];       // [101]
  const float* b = (const float*)d_in[2];       // [1]
  const float* points = (const float*)d_in[3];  // [100]
  float* out = (float*)d_out;

  const int n = in_sizes[0];

  float* ws = (float*)d_ws;
  float* k_tab = ws;        // 101 floats
  float* dbias = ws + 128;  // 100 floats (aligned slot)

  pwl_prep_kernel<<<1, 128, 0, stream>>>(p, b, k_tab, dbias);

  int n4 = n / 4;
  int blocks = 2048;
  int need = (n4 + 255) / 256;
  if (need < 1) need = 1;
  if (blocks > need) blocks = need;

  pwl_main_kernel<<<blocks, 256, 0, stream>>>(eps, points, k_tab, dbias, out, n,
                                              n4);
}